// Fusion_45578192945950
// MI455X (gfx1250) — compile-verified
//
#include <hip/hip_runtime.h>
#include <hip/hip_bf16.h>

// ---------------------------------------------------------------------------
// Sizes from the reference
// ---------------------------------------------------------------------------
#define HID   1024
#define ATT   256
#define MROWS 768   // M1 == M2 == 768

typedef __attribute__((ext_vector_type(2))) float v2f;
typedef __attribute__((ext_vector_type(8))) float v8f;

// ---------------------------------------------------------------------------
// Hardware tanh (CDNA5 v_tanh_f32). Builtin if available, else inline asm.
// The trailing v_nop in the asm fallback satisfies the TRANS-op co-execution
// hazard (1 independent VALU op before the result may be consumed).
// ---------------------------------------------------------------------------
__device__ __forceinline__ float hw_tanh(float x) {
#if __has_builtin(__builtin_amdgcn_tanhf)
  return __builtin_amdgcn_tanhf(x);
#else
  float y;
  asm volatile("v_tanh_f32 %0, %1\n\tv_nop" : "=v"(y) : "v"(x));
  return y;
#endif
}

// ---------------------------------------------------------------------------
// Kernel A: x = h @ W.T + bias via V_WMMA_F32_16X16X4_F32.
// One wave (32 threads) per 16x16 output tile. K-loop steps of 4.
//
// Fragment layouts (ISA 7.12.2, 32-bit 16x4 A / 4x16 B):
//   A: lane l<16 holds {A[l,k0],A[l,k0+1]}, lane l>=16 holds {A[l-16,k0+2],A[l-16,k0+3]}
//   B: lane l<16 holds {B[k0,n0+l],B[k0+1,n0+l]} = {W[n0+l,k0],W[n0+l,k0+1]}, upper half k0+2/3
// so each lane streams a float2 from one h-row (A) and one W-row (B).
//
// blockIdx.y selects (h1,W1,b1,x1) vs (h2,W2,b2,x2).
// The first 48 blocks of y==0 also zero the rowsum/colsum accumulators.
// ---------------------------------------------------------------------------
__global__ void __launch_bounds__(32)
gemm_proj_kernel(const float* __restrict__ h1, const float* __restrict__ W1,
                 const float* __restrict__ b1, const float* __restrict__ h2,
                 const float* __restrict__ W2, const float* __restrict__ b2,
                 float* __restrict__ x1, float* __restrict__ x2,
                 float* __restrict__ rowcol /* 1536 floats to zero */) {
  const int tile = blockIdx.x;            // 0..767  (48 M-tiles x 16 N-tiles)
  const int lane = threadIdx.x;           // 0..31
  const int m0 = (tile >> 4) << 4;        // M tile base
  const int n0 = (tile & 15) << 4;        // N tile base
  const int half = lane >> 4;             // 0 | 1
  const int l = lane & 15;

  if (blockIdx.y == 0 && tile < 48) rowcol[tile * 32 + lane] = 0.0f;

  const float* __restrict__ H = (blockIdx.y == 0) ? h1 : h2;
  const float* __restrict__ W = (blockIdx.y == 0) ? W1 : W2;
  const float* __restrict__ B = (blockIdx.y == 0) ? b1 : b2;
  float* __restrict__ X       = (blockIdx.y == 0) ? x1 : x2;

  const float* arow = H + (size_t)(m0 + l) * HID + 2 * half;
  const float* brow = W + (size_t)(n0 + l) * HID + 2 * half;

  v8f c = {0.f, 0.f, 0.f, 0.f, 0.f, 0.f, 0.f, 0.f};
#pragma unroll 8
  for (int k = 0; k < HID; k += 4) {
    v2f a = *(const v2f*)(arow + k);
    v2f b = *(const v2f*)(brow + k);
    c = __builtin_amdgcn_wmma_f32_16x16x4_f32(
        /*neg_a=*/false, a, /*neg_b=*/false, b,
        /*c_mod=*/(short)0, c, /*reuse_a=*/false, /*reuse_b=*/false);
  }

  // C/D layout: VGPR i -> row m0+i (lanes 0-15) / m0+i+8 (lanes 16-31), col n0+(lane&15)
  const float bv = B[n0 + l];
#pragma unroll
  for (int i = 0; i < 8; ++i) {
    const int m = m0 + i + 8 * half;
    X[(size_t)m * ATT + n0 + l] = c[i] + bv;
  }
}

// ---------------------------------------------------------------------------
// Kernel B: rowsum[m] += sum_n alpha[m,n], colsum[n] += sum_m alpha[m,n]
// where alpha[m,n] = sum_a w[a] * tanh(x1[m,a] + x2[n,a]).
//
// 64x64 (m,n) tile per block, 256 threads, 4x4 register blocking per thread.
// Tiles staged in LDS transposed as [a][m] with stride 68 (bank-conflict-free
// reads: lanes with distinct tm hit distinct banks; same-address lanes use
// LDS broadcast). ~137 KB static LDS — fits CDNA5's 320 KB/WGP.
// ---------------------------------------------------------------------------
#define TSTRIDE 68

__global__ void __launch_bounds__(256)
alpha_reduce_kernel(const float* __restrict__ X1, const float* __restrict__ X2,
                    const float* __restrict__ w,
                    float* __restrict__ rowsum, float* __restrict__ colsum) {
  __shared__ float s1t[ATT * TSTRIDE];  // [a][m], 69,632 B
  __shared__ float s2t[ATT * TSTRIDE];  // [a][n]
  __shared__ float sw[ATT];

  const int tid = threadIdx.x;
  const int m0 = blockIdx.x * 64;
  const int n0 = blockIdx.y * 64;

  // Stage tiles (transposed) + w into LDS
  for (int i = tid; i < 64 * 64; i += 256) {
    const int r = i >> 6;              // local row (m or n), 0..63
    const int c = (i & 63) << 2;       // a base, 0..252 step 4
    const float4 v1 = *(const float4*)&X1[(size_t)(m0 + r) * ATT + c];
    const float4 v2 = *(const float4*)&X2[(size_t)(n0 + r) * ATT + c];
    s1t[(c + 0) * TSTRIDE + r] = v1.x; s1t[(c + 1) * TSTRIDE + r] = v1.y;
    s1t[(c + 2) * TSTRIDE + r] = v1.z; s1t[(c + 3) * TSTRIDE + r] = v1.w;
    s2t[(c + 0) * TSTRIDE + r] = v2.x; s2t[(c + 1) * TSTRIDE + r] = v2.y;
    s2t[(c + 2) * TSTRIDE + r] = v2.z; s2t[(c + 3) * TSTRIDE + r] = v2.w;
  }
  if (tid < 64) *(float4*)&sw[tid * 4] = *(const float4*)&w[tid * 4];
  __syncthreads();

  const int tm = tid & 15;   // m sub-tile 0..15 (4 rows each)
  const int tn = tid >> 4;   // n sub-tile 0..15 (4 cols each)

  float acc[4][4];
#pragma unroll
  for (int r = 0; r < 4; ++r)
#pragma unroll
    for (int cc = 0; cc < 4; ++cc) acc[r][cc] = 0.f;

#pragma unroll 4
  for (int a = 0; a < ATT; ++a) {
    const float wa = sw[a];
    float av[4], bv[4];
#pragma unroll
    for (int r = 0; r < 4; ++r)  av[r] = s1t[a * TSTRIDE + 4 * tm + r];
#pragma unroll
    for (int cc = 0; cc < 4; ++cc) bv[cc] = s2t[a * TSTRIDE + 4 * tn + cc];
#pragma unroll
    for (int r = 0; r < 4; ++r)
#pragma unroll
      for (int cc = 0; cc < 4; ++cc)
        acc[r][cc] = fmaf(wa, hw_tanh(av[r] + bv[cc]), acc[r][cc]);
  }

  // Per-thread partials
  float rpart[4], cpart[4];
#pragma unroll
  for (int r = 0; r < 4; ++r) {
    rpart[r] = acc[r][0] + acc[r][1] + acc[r][2] + acc[r][3];
  }
#pragma unroll
  for (int cc = 0; cc < 4; ++cc) {
    cpart[cc] = acc[0][cc] + acc[1][cc] + acc[2][cc] + acc[3][cc];
  }

  __syncthreads();                 // done reading tiles; reuse s1t as scratch
  float* red = s1t;                // [0..1023]: rows, [1024..2047]: cols
#pragma unroll
  for (int r = 0; r < 4; ++r)   red[(4 * tm + r) * 16 + tn] = rpart[r];
#pragma unroll
  for (int cc = 0; cc < 4; ++cc) red[1024 + (4 * tn + cc) * 16 + tm] = cpart[cc];
  __syncthreads();

  if (tid < 64) {
    float s = 0.f;
#pragma unroll
    for (int j = 0; j < 16; ++j) s += red[tid * 16 + j];
    atomicAdd(&rowsum[m0 + tid], s);
  } else if (tid < 128) {
    const int n = tid - 64;
    float s = 0.f;
#pragma unroll
    for (int j = 0; j < 16; ++j) s += red[1024 + n * 16 + j];
    atomicAdd(&colsum[n0 + n], s);
  }
}

// ---------------------------------------------------------------------------
// Kernel C: softmax(mean) for both axes + s = X.T @ p. One block, 256 threads.
// out[0:256) = x1.T @ softmax(rowsum/768), out[256:512) = x2.T @ softmax(colsum/768)
// ---------------------------------------------------------------------------
__device__ __forceinline__ float blk_reduce(float v, float* red, int tid, bool is_max) {
  red[tid] = v;
  __syncthreads();
  for (int o = 128; o > 0; o >>= 1) {
    if (tid < o) red[tid] = is_max ? fmaxf(red[tid], red[tid + o])
                                   : (red[tid] + red[tid + o]);
    __syncthreads();
  }
  const float r = red[0];
  __syncthreads();
  return r;
}

__global__ void __launch_bounds__(256)
finalize_kernel(const float* __restrict__ X1, const float* __restrict__ X2,
                const float* __restrict__ rowsum, const float* __restrict__ colsum,
                float* __restrict__ out) {
  __shared__ float p[MROWS];
  __shared__ float red[256];
  const int tid = threadIdx.x;
  const float inv = 1.0f / (float)MROWS;

  for (int side = 0; side < 2; ++side) {
    const float* acc = (side == 0) ? rowsum : colsum;
    const float* X   = (side == 0) ? X1 : X2;

    const float v0 = acc[tid] * inv;
    const float v1 = acc[tid + 256] * inv;
    const float v2 = acc[tid + 512] * inv;
    const float mx = blk_reduce(fmaxf(v0, fmaxf(v1, v2)), red, tid, true);
    const float e0 = __expf(v0 - mx);
    const float e1 = __expf(v1 - mx);
    const float e2 = __expf(v2 - mx);
    const float sum = blk_reduce(e0 + e1 + e2, red, tid, false);
    const float isum = 1.0f / sum;
    p[tid] = e0 * isum;
    p[tid + 256] = e1 * isum;
    p[tid + 512] = e2 * isum;
    __syncthreads();

    float s = 0.f;
    for (int m = 0; m < MROWS; ++m)
      s = fmaf(X[(size_t)m * ATT + tid], p[m], s);  // p[m]: LDS broadcast
    out[side * ATT + tid] = s;
    __syncthreads();
  }
}

// ---------------------------------------------------------------------------
// Launch. Inputs: h1, h2, W1, bias1, W2, bias2, w (all f32). out: 512 f32.
// Workspace: x1 (768*256) | x2 (768*256) | rowsum (768) | colsum (768)
// = ~1.58 MB of f32.
// ---------------------------------------------------------------------------
extern "C" void kernel_launch(void* const* d_in, const int* in_sizes, int n_in,
                              void* d_out, int out_size, void* d_ws, size_t ws_size,
                              hipStream_t stream) {
  const float* h1 = (const float*)d_in[0];
  const float* h2 = (const float*)d_in[1];
  const float* W1 = (const float*)d_in[2];
  const float* b1 = (const float*)d_in[3];
  const float* W2 = (const float*)d_in[4];
  const float* b2 = (const float*)d_in[5];
  const float* w  = (const float*)d_in[6];
  float* out = (float*)d_out;

  float* x1     = (float*)d_ws;
  float* x2     = x1 + (size_t)MROWS * ATT;
  float* rowsum = x2 + (size_t)MROWS * ATT;
  float* colsum = rowsum + MROWS;
  // rowsum/colsum are contiguous: 1536 floats, zeroed inside gemm_proj_kernel.

  // A) projections via f32 WMMA (+ accumulator zeroing)
  gemm_proj_kernel<<<dim3(768, 2), dim3(32), 0, stream>>>(
      h1, W1, b1, h2, W2, b2, x1, x2, rowsum);

  // B) tanh row/col reduction
  alpha_reduce_kernel<<<dim3(12, 12), dim3(256), 0, stream>>>(
      x1, x2, w, rowsum, colsum);

  // C) softmaxes + weighted sums
  finalize_kernel<<<dim3(1), dim3(256), 0, stream>>>(x1, x2, rowsum, colsum, out);
}